// Block_41893111005237
// MI455X (gfx1250) — compile-verified
//
#include <hip/hip_runtime.h>
#include <math.h>
#include <stdint.h>

// ---- problem constants (from reference) ----
#define D_MODEL_  1024
#define NH_ATT    16
#define HD_       64
#define D_STATE_  128
#define D_INNER_  2048
#define NHEADS_   32
#define HEADDIM_  64
#define FFN_      4096
#define BSZ_      2
#define TN_       1024
#define BT_       (BSZ_*TN_)
#define ZDIM_     (2*D_INNER_ + 2*D_STATE_ + NHEADS_)   // 4384
#define XBC_      (D_INNER_ + 2*D_STATE_)               // 2304

typedef __attribute__((ext_vector_type(2))) float v2f;
typedef __attribute__((ext_vector_type(8))) float v8f;

// ---------------- block reduce (wave32, 256 threads) ----------------
__device__ __forceinline__ float block_reduce_sum(float v) {
  for (int o = 16; o > 0; o >>= 1) v += __shfl_xor(v, o, 32);
  __shared__ float s[8];
  int lane = threadIdx.x & 31, w = threadIdx.x >> 5;
  if (lane == 0) s[w] = v;
  __syncthreads();
  float r = (threadIdx.x < 8) ? s[threadIdx.x] : 0.f;
  if (w == 0) {
    for (int o = 4; o > 0; o >>= 1) r += __shfl_xor(r, o, 32);
    if (lane == 0) s[0] = r;
  }
  __syncthreads();
  return s[0];
}

// ---------------- RMSNorm: one block per row ----------------
__global__ void rmsnorm_kernel(const float* __restrict__ x, float* __restrict__ y,
                               int D, float eps) {
  size_t row = blockIdx.x;
  const float* xr = x + row * D;
  float* yr = y + row * D;
  float ss = 0.f;
  for (int c = threadIdx.x; c < D; c += 256) { float v = xr[c]; ss += v * v; }
  ss = block_reduce_sum(ss);
  float rs = rsqrtf(ss / (float)D + eps);
  for (int c = threadIdx.x; c < D; c += 256) yr[c] = xr[c] * rs;
}

// ---------------- WMMA f32 GEMM: D = A[M,K] x B[K,N] (+Cadd) ----------------
// Block = 4 waves; wave w computes a 64x16 tile at (blockIdx.y*256 + w*64, blockIdx.x*16).
// Per K-step of 4 the wave loads ONE B fragment and four A fragments and issues
// four V_WMMA_F32_16X16X4_F32 (4x B reuse vs the naive tiling).
// A frag (16x4): lanes 0-15 hold K=k0+0,k0+1 for row M=lane; lanes 16-31 K=k0+2,k0+3.
// B frag (4x16): lanes 0-15 hold rows k0+0,k0+1 at col N=lane; lanes 16-31 rows k0+2,k0+3.
// C/D: VGPR r, lane<16 -> (M=r, N=lane); lane>=16 -> (M=8+r, N=lane-16).
// act: 0 = none, 1 = relu(x)^2
__global__ void wmma_gemm_f32(const float* __restrict__ A, const float* __restrict__ B,
                              const float* __restrict__ Cadd, float* __restrict__ D,
                              int M, int N, int K, int act) {
  int tn   = blockIdx.x * 16;
  int wave = threadIdx.x >> 5;
  int tm   = blockIdx.y * 256 + wave * 64;
  int lane = threadIdx.x & 31;
  int half = lane >> 4;
  int l15  = lane & 15;

  v8f acc[4];
  acc[0] = v8f{}; acc[1] = v8f{}; acc[2] = v8f{}; acc[3] = v8f{};

  const float* ap0 = A + (size_t)(tm + l15) * K + 2 * half;
  const float* ap1 = ap0 + (size_t)16 * K;
  const float* ap2 = ap0 + (size_t)32 * K;
  const float* ap3 = ap0 + (size_t)48 * K;
  const float* bp  = B + (size_t)(2 * half) * N + tn + l15;

  for (int k0 = 0; k0 < K; k0 += 4) {
    v2f b;  b.x  = bp[0];  b.y  = bp[N];
    v2f a0; a0.x = ap0[0]; a0.y = ap0[1];
    v2f a1; a1.x = ap1[0]; a1.y = ap1[1];
    v2f a2; a2.x = ap2[0]; a2.y = ap2[1];
    v2f a3; a3.x = ap3[0]; a3.y = ap3[1];
    acc[0] = __builtin_amdgcn_wmma_f32_16x16x4_f32(false, a0, false, b, (short)0, acc[0], false, false);
    acc[1] = __builtin_amdgcn_wmma_f32_16x16x4_f32(false, a1, false, b, (short)0, acc[1], false, false);
    acc[2] = __builtin_amdgcn_wmma_f32_16x16x4_f32(false, a2, false, b, (short)0, acc[2], false, false);
    acc[3] = __builtin_amdgcn_wmma_f32_16x16x4_f32(false, a3, false, b, (short)0, acc[3], false, false);
    ap0 += 4; ap1 += 4; ap2 += 4; ap3 += 4;
    bp  += (size_t)4 * N;
  }

  int col = tn + l15;
  bool hasC = (Cadd != nullptr);
  #pragma unroll
  for (int m = 0; m < 4; ++m) {
    int rbase = tm + m * 16 + half * 8;
    #pragma unroll
    for (int r = 0; r < 8; ++r) {
      int row = rbase + r;
      float v = acc[m][r];
      if (hasC) v += Cadd[(size_t)row * N + col];
      if (act == 1) v = (v > 0.f) ? v * v : 0.f;
      D[(size_t)row * N + col] = v;
    }
  }
}

// ---------------- causal depthwise conv (K=4) + SiLU on xBC ----------------
__global__ void conv_silu_kernel(const float* __restrict__ zx, const float* __restrict__ w,
                                 const float* __restrict__ bias, float* __restrict__ out) {
  int idx = blockIdx.x * 256 + threadIdx.x;
  if (idx >= BT_ * XBC_) return;
  int c = idx % XBC_;
  int bt = idx / XBC_;
  int t = bt % TN_, b = bt / TN_;
  const float* base = zx + (size_t)(b * TN_) * ZDIM_ + D_INNER_ + c;
  float acc = bias[c];
  #pragma unroll
  for (int k = 0; k < 4; ++k) {
    int tt = t - 3 + k;
    if (tt >= 0) acc += w[c * 4 + k] * base[(size_t)tt * ZDIM_];
  }
  float s = acc / (1.f + __expf(-acc));
  out[(size_t)bt * XBC_ + c] = s;
}

// ---------------- dt = softplus(dt + bias) ----------------
__global__ void dt_softplus_kernel(const float* __restrict__ zx,
                                   const float* __restrict__ dt_bias,
                                   float* __restrict__ dt_out) {
  int idx = blockIdx.x * 256 + threadIdx.x;
  if (idx >= BT_ * NHEADS_) return;
  int h = idx % NHEADS_;
  int bt = idx / NHEADS_;
  float v = zx[(size_t)bt * ZDIM_ + (ZDIM_ - NHEADS_) + h] + dt_bias[h];
  dt_out[idx] = (v > 20.f) ? v : log1pf(__expf(v));
}

// ---------------- SSM scan: one workgroup per (batch, head) ----------------
// 256 threads: thread t -> p = t>>2 (0..63), quarter q = t&3 owns 32 n's of state.
// B_t (128 f32) and C_t (128 f32) are contiguous in xbc; they are double-buffered
// into LDS with GLOBAL_LOAD_ASYNC_TO_LDS_B32 (ASYNCcnt) so the step s+1 fetch
// overlaps the 64-FMA state update of step s. Async loads complete in order, so
// s_wait_asynccnt 1 guarantees the current step's buffer has landed.
__global__ void ssm_scan_kernel(const float* __restrict__ xbc, const float* __restrict__ dt,
                                const float* __restrict__ A_log, float* __restrict__ ys) {
  int bh = blockIdx.x;
  int b = bh >> 5, h = bh & 31;
  int t = threadIdx.x;
  int p = t >> 2, q = t & 3;
  float A = -__expf(A_log[h]);
  float state[32];
  #pragma unroll
  for (int i = 0; i < 32; ++i) state[i] = 0.f;

  __shared__ float sBC[2][2 * D_STATE_];   // [buffer][B(0..127) C(128..255)]
  // element t of a step's staging block lives at xbc[row*XBC_ + D_INNER_ + t]
  const float* gsrc = xbc + (size_t)(b * TN_) * XBC_ + D_INNER_ + t;
  uint32_t lds0 = (uint32_t)(uintptr_t)&sBC[0][t];
  uint32_t lds1 = (uint32_t)(uintptr_t)&sBC[1][t];

  // prefetch step 0 into buffer 0
  asm volatile("global_load_async_to_lds_b32 %0, %1, off"
               :: "v"(lds0), "v"(gsrc) : "memory");

  for (int step = 0; step < TN_; ++step) {
    int cur = step & 1;
    if (step + 1 < TN_) {
      const float* gn = gsrc + (size_t)(step + 1) * XBC_;
      uint32_t ldsn = cur ? lds0 : lds1;       // buffer (cur^1)
      asm volatile("global_load_async_to_lds_b32 %0, %1, off"
                   :: "v"(ldsn), "v"(gn) : "memory");
      asm volatile("s_wait_asynccnt 0x1" ::: "memory");
    } else {
      asm volatile("s_wait_asynccnt 0x0" ::: "memory");
    }
    __syncthreads();                           // current buffer visible to all waves

    size_t row = (size_t)(b * TN_ + step);
    float dtv = dt[row * NHEADS_ + h];
    float dA  = __expf(dtv * A);
    float xv  = xbc[row * XBC_ + h * HEADDIM_ + p];
    float dx  = dtv * xv;
    float partial = 0.f;
    const float* Bq = &sBC[cur][q * 32];
    const float* Cq = &sBC[cur][D_STATE_ + q * 32];
    #pragma unroll
    for (int i = 0; i < 32; ++i) {
      state[i] = state[i] * dA + dx * Bq[i];
      partial += state[i] * Cq[i];
    }
    partial += __shfl_xor(partial, 1, 32);
    partial += __shfl_xor(partial, 2, 32);
    if (q == 0) ys[row * D_INNER_ + h * HEADDIM_ + p] = partial;
    __syncthreads();                           // done reading before next overwrite
  }
}

// ---------------- gate: (ys + x*Dm)*silu(z) then RMSNorm*mnorm_w ----------------
__global__ void gate_norm_kernel(const float* __restrict__ ys, const float* __restrict__ xbc,
                                 const float* __restrict__ zx, const float* __restrict__ Dm,
                                 const float* __restrict__ mnorm_w, float* __restrict__ yn) {
  size_t row = blockIdx.x;
  float u[8];
  float ss = 0.f;
  #pragma unroll
  for (int i = 0; i < 8; ++i) {
    int c = threadIdx.x + i * 256;
    int h = c >> 6;
    float xv = xbc[row * XBC_ + c];
    float yv = ys[row * D_INNER_ + c] + xv * Dm[h];
    float z  = zx[row * ZDIM_ + c];
    float sz = z / (1.f + __expf(-z));
    float val = yv * sz;
    u[i] = val;
    ss += val * val;
  }
  ss = block_reduce_sum(ss);
  float rs = rsqrtf(ss / (float)D_INNER_ + 1e-5f);
  #pragma unroll
  for (int i = 0; i < 8; ++i) {
    int c = threadIdx.x + i * 256;
    yn[row * D_INNER_ + c] = u[i] * rs * mnorm_w[c];
  }
}

// ---------------- per-dim causal conv (K=3) on q/k/v ----------------
__global__ void qkv_conv_kernel(const float* __restrict__ qkv,
                                const float* __restrict__ qw, const float* __restrict__ qb,
                                const float* __restrict__ kw, const float* __restrict__ kb,
                                const float* __restrict__ vw, const float* __restrict__ vb,
                                float* __restrict__ qc, float* __restrict__ kc,
                                float* __restrict__ vc) {
  int idx = blockIdx.x * 256 + threadIdx.x;
  if (idx >= BT_ * 1152) return;
  int col = idx % 1152;
  int bt = idx / 1152;
  int t = bt % TN_, b = bt / TN_;
  const float* W; const float* Bb; int d; float* dst;
  if (col < 1024)      { d = col & 63;   W = qw; Bb = qb; dst = qc + (size_t)bt * 1024 + col; }
  else if (col < 1088) { d = col - 1024; W = kw; Bb = kb; dst = kc + (size_t)bt * 64 + d; }
  else                 { d = col - 1088; W = vw; Bb = vb; dst = vc + (size_t)bt * 64 + d; }
  const float* base = qkv + (size_t)(b * TN_) * 1152 + col;
  float acc = Bb[d];
  #pragma unroll
  for (int k = 0; k < 3; ++k) {
    int tt = t - 2 + k;
    if (tt >= 0) acc += W[d * 3 + k] * base[(size_t)tt * 1152];
  }
  *dst = acc;
}

// ---------------- causal MQA attention, one wave per query row ----------------
__global__ void attn_kernel(const float* __restrict__ qc, const float* __restrict__ kc,
                            const float* __restrict__ vc, float* __restrict__ yatt) {
  int gw = blockIdx.x * 8 + (threadIdx.x >> 5);   // global wave id over B*H*T = 32768
  int lane = threadIdx.x & 31;
  if (gw >= BSZ_ * NH_ATT * TN_) return;
  int tq = gw % TN_;
  int h  = (gw / TN_) % NH_ATT;
  int b  = gw / (TN_ * NH_ATT);
  const float* qrow = qc + (size_t)(b * TN_ + tq) * 1024 + h * 64;
  float q0 = qrow[lane], q1 = qrow[lane + 32];
  float m = -1e30f, l = 0.f, a0 = 0.f, a1 = 0.f;
  const float* kbase = kc + (size_t)(b * TN_) * 64;
  const float* vbase = vc + (size_t)(b * TN_) * 64;
  for (int s = 0; s <= tq; ++s) {
    float k0 = kbase[s * 64 + lane], k1 = kbase[s * 64 + lane + 32];
    float pd = q0 * k0 + q1 * k1;
    pd += __shfl_xor(pd, 16, 32);
    pd += __shfl_xor(pd, 8, 32);
    pd += __shfl_xor(pd, 4, 32);
    pd += __shfl_xor(pd, 2, 32);
    pd += __shfl_xor(pd, 1, 32);
    float score = pd * 0.125f;                 // / sqrt(64)
    float mn = fmaxf(m, score);
    float sc = __expf(m - mn);
    float p  = __expf(score - mn);
    l = l * sc + p;
    float v0 = vbase[s * 64 + lane], v1 = vbase[s * 64 + lane + 32];
    a0 = a0 * sc + p * v0;
    a1 = a1 * sc + p * v1;
    m = mn;
  }
  float inv = 1.f / l;
  float* o = yatt + (size_t)(b * TN_ + tq) * 1024 + h * 64;
  o[lane] = a0 * inv;
  o[lane + 32] = a1 * inv;
}

// ---------------- RWKV token shift ----------------
__global__ void tokenshift_kernel(const float* __restrict__ h3,
                                  const float* __restrict__ maa_k,
                                  const float* __restrict__ maa_r,
                                  float* __restrict__ xk, float* __restrict__ xr) {
  int idx = blockIdx.x * 256 + threadIdx.x;
  if (idx >= BT_ * D_MODEL_) return;
  int c = idx % D_MODEL_;
  int bt = idx / D_MODEL_;
  int t = bt % TN_;
  float hv = h3[idx];
  float prev = (t == 0) ? 0.f : h3[idx - D_MODEL_];
  float xx = prev - hv;
  xk[idx] = hv + xx * maa_k[c];
  xr[idx] = hv + xx * maa_r[c];
}

// ---------------- final: out = x3 + sigmoid(r)*kv ----------------
__global__ void final_kernel(const float* __restrict__ x3, const float* __restrict__ r,
                             const float* __restrict__ kv, float* __restrict__ out) {
  int idx = blockIdx.x * 256 + threadIdx.x;
  if (idx >= BT_ * D_MODEL_) return;
  out[idx] = x3[idx] + kv[idx] / (1.f + __expf(-r[idx]));
}

// ============================================================================
extern "C" void kernel_launch(void* const* d_in, const int* in_sizes, int n_in,
                              void* d_out, int out_size, void* d_ws, size_t ws_size,
                              hipStream_t stream) {
  const float* x        = (const float*)d_in[0];
  const float* W_in     = (const float*)d_in[1];
  const float* conv_w   = (const float*)d_in[2];
  const float* conv_b   = (const float*)d_in[3];
  const float* A_log    = (const float*)d_in[4];
  const float* Dm       = (const float*)d_in[5];
  const float* dt_bias  = (const float*)d_in[6];
  const float* mnorm_w  = (const float*)d_in[7];
  const float* W_out    = (const float*)d_in[8];
  const float* W_qkv    = (const float*)d_in[9];
  const float* W_cproj  = (const float*)d_in[10];
  const float* qconv_w  = (const float*)d_in[11];
  const float* qconv_b  = (const float*)d_in[12];
  const float* kconv_w  = (const float*)d_in[13];
  const float* kconv_b  = (const float*)d_in[14];
  const float* vconv_w  = (const float*)d_in[15];
  const float* vconv_b  = (const float*)d_in[16];
  const float* maa_k    = (const float*)d_in[17];
  const float* maa_r    = (const float*)d_in[18];
  const float* W_key    = (const float*)d_in[19];
  const float* W_rec    = (const float*)d_in[20];
  const float* W_val    = (const float*)d_in[21];

  float* ws = (float*)d_ws;
  float* h1   = ws; ws += (size_t)BT_ * D_MODEL_;
  float* zx   = ws; ws += (size_t)BT_ * ZDIM_;
  float* xbc  = ws; ws += (size_t)BT_ * XBC_;
  float* dta  = ws; ws += (size_t)BT_ * NHEADS_;
  float* ys   = ws; ws += (size_t)BT_ * D_INNER_;
  float* yn   = ws; ws += (size_t)BT_ * D_INNER_;
  float* x2   = ws; ws += (size_t)BT_ * D_MODEL_;
  float* h2   = ws; ws += (size_t)BT_ * D_MODEL_;
  float* qkv  = ws; ws += (size_t)BT_ * 1152;
  float* qc   = ws; ws += (size_t)BT_ * 1024;
  float* kc   = ws; ws += (size_t)BT_ * 64;
  float* vc   = ws; ws += (size_t)BT_ * 64;
  float* yatt = ws; ws += (size_t)BT_ * D_MODEL_;
  float* x3   = ws; ws += (size_t)BT_ * D_MODEL_;
  float* h3   = ws; ws += (size_t)BT_ * D_MODEL_;
  float* xk   = ws; ws += (size_t)BT_ * D_MODEL_;
  float* xr   = ws; ws += (size_t)BT_ * D_MODEL_;
  float* kf   = ws; ws += (size_t)BT_ * FFN_;
  float* kv   = ws; ws += (size_t)BT_ * D_MODEL_;
  float* rr   = ws; ws += (size_t)BT_ * D_MODEL_;

  dim3 b256(256);
  dim3 b128(128);   // 4 waves per GEMM block, each wave a 64x16 tile

  // --- Mamba branch ---
  rmsnorm_kernel<<<BT_, b256, 0, stream>>>(x, h1, D_MODEL_, 1e-6f);
  wmma_gemm_f32<<<dim3(ZDIM_ / 16, BT_ / 256), b128, 0, stream>>>(h1, W_in, nullptr, zx,
                                                                  BT_, ZDIM_, D_MODEL_, 0);
  conv_silu_kernel<<<(BT_ * XBC_) / 256, b256, 0, stream>>>(zx, conv_w, conv_b, xbc);
  dt_softplus_kernel<<<(BT_ * NHEADS_) / 256, b256, 0, stream>>>(zx, dt_bias, dta);
  ssm_scan_kernel<<<BSZ_ * NHEADS_, b256, 0, stream>>>(xbc, dta, A_log, ys);
  gate_norm_kernel<<<BT_, b256, 0, stream>>>(ys, xbc, zx, Dm, mnorm_w, yn);
  wmma_gemm_f32<<<dim3(D_MODEL_ / 16, BT_ / 256), b128, 0, stream>>>(yn, W_out, x, x2,
                                                                     BT_, D_MODEL_, D_INNER_, 0);

  // --- attention branch ---
  rmsnorm_kernel<<<BT_, b256, 0, stream>>>(x2, h2, D_MODEL_, 1e-6f);
  wmma_gemm_f32<<<dim3(1152 / 16, BT_ / 256), b128, 0, stream>>>(h2, W_qkv, nullptr, qkv,
                                                                 BT_, 1152, D_MODEL_, 0);
  qkv_conv_kernel<<<(BT_ * 1152) / 256, b256, 0, stream>>>(qkv, qconv_w, qconv_b,
                                                           kconv_w, kconv_b,
                                                           vconv_w, vconv_b, qc, kc, vc);
  attn_kernel<<<(BSZ_ * NH_ATT * TN_) / 8, b256, 0, stream>>>(qc, kc, vc, yatt);
  wmma_gemm_f32<<<dim3(D_MODEL_ / 16, BT_ / 256), b128, 0, stream>>>(yatt, W_cproj, x2, x3,
                                                                     BT_, D_MODEL_, D_MODEL_, 0);

  // --- RWKV-ish channel mix ---
  rmsnorm_kernel<<<BT_, b256, 0, stream>>>(x3, h3, D_MODEL_, 1e-6f);
  tokenshift_kernel<<<(BT_ * D_MODEL_) / 256, b256, 0, stream>>>(h3, maa_k, maa_r, xk, xr);
  wmma_gemm_f32<<<dim3(FFN_ / 16, BT_ / 256), b128, 0, stream>>>(xk, W_key, nullptr, kf,
                                                                 BT_, FFN_, D_MODEL_, 1); // relu^2
  wmma_gemm_f32<<<dim3(D_MODEL_ / 16, BT_ / 256), b128, 0, stream>>>(kf, W_val, nullptr, kv,
                                                                     BT_, D_MODEL_, FFN_, 0);
  wmma_gemm_f32<<<dim3(D_MODEL_ / 16, BT_ / 256), b128, 0, stream>>>(xr, W_rec, nullptr, rr,
                                                                     BT_, D_MODEL_, D_MODEL_, 0);
  final_kernel<<<(BT_ * D_MODEL_) / 256, b256, 0, stream>>>(x3, rr, kv, (float*)d_out);
}